// UCBNorm_41308995453348
// MI455X (gfx1250) — compile-verified
//
#include <hip/hip_runtime.h>
#include <math.h>

// ---------------------------------------------------------------------------
// UCBNorm for MI455X (gfx1250, wave32).
// Two streaming passes over x + tiny reductions. The 16-way t-chunk partial
// reduction uses V_WMMA_F32_16X16X4_F32 with a ones A-matrix (f32-exact).
// Streaming loops use global_prefetch_b8 via __builtin_prefetch.
// ---------------------------------------------------------------------------

typedef __attribute__((ext_vector_type(2))) float v2f;
typedef __attribute__((ext_vector_type(8))) float v8f;

namespace {
constexpr int   B_  = 16;
constexpr int   T_  = 2048;
constexpr int   D_  = 256;
constexpr int   K_  = 8;
constexpr int   TC_ = 16;          // t-chunks per batch row
constexpr int   CT_ = T_ / TC_;    // 128 timesteps per chunk
constexpr float EPS_   = 1e-3f;
constexpr float LOG2E_ = 1.4426950408889634f;
}

// ---- kernel 1: per (k,d) exp2-form Gaussian coefficient --------------------
// c1 = -0.5 * log2(e) / (softplus(variance) + eps)   so p = exp2(c1 * dd*dd)
__global__ void k_precompute(const float* __restrict__ var,
                             float* __restrict__ c1) {
    int i = blockIdx.x * blockDim.x + threadIdx.x;     // < K*D
    float v  = var[i];
    float sp = fmaxf(v, 0.0f) + log1pf(__expf(-fabsf(v)));  // stable softplus
    c1[i] = -0.5f * LOG2E_ / (sp + EPS_);
}

// ---- kernel 2: streaming pass A -------------------------------------------
// thread = d (256), block = (t-chunk, b). Accumulates per-thread:
//   S_k  = sum_t tau            (partial of sum over all b,t)
//   A1_k = sum_t tau * x
//   A3_k = sum_t tau^3 * x^2
__global__ void k_passA(const float* __restrict__ x,
                        const float* __restrict__ mean,
                        const float* __restrict__ c1,
                        float* __restrict__ Spart,   // [B*TC][K][D]
                        float* __restrict__ A1p,     // [K][B][TC][D]
                        float* __restrict__ A3p) {   // [K][B][TC][D]
    const int d  = threadIdx.x;
    const int tc = blockIdx.x;
    const int b  = blockIdx.y;

    float mk[K_], ck[K_], aS[K_], a1[K_], a3[K_];
#pragma unroll
    for (int k = 0; k < K_; ++k) {
        mk[k] = mean[k * D_ + d];
        ck[k] = c1[k * D_ + d];
        aS[k] = 0.0f; a1[k] = 0.0f; a3[k] = 0.0f;
    }

    const size_t total = (size_t)B_ * T_ * D_;
    const size_t base  = ((size_t)b * T_ + (size_t)tc * CT_) * D_ + d;

    for (int t = 0; t < CT_; ++t) {
        const size_t idx = base + (size_t)t * D_;
        const size_t pf  = idx + 16 * (size_t)D_;
        if (pf < total) __builtin_prefetch(x + pf, 0, 0);   // global_prefetch_b8

        const float xv = x[idx];
        float p[K_];
        float s = 0.0f;
#pragma unroll
        for (int k = 0; k < K_; ++k) {
            const float dd = xv - mk[k];
            p[k] = __builtin_amdgcn_exp2f(ck[k] * dd * dd); // v_exp_f32
            s += p[k];
        }
        const float inv = __builtin_amdgcn_rcpf(s + EPS_);
        const float x2  = xv * xv;
#pragma unroll
        for (int k = 0; k < K_; ++k) {
            const float tau = p[k] * inv;
            aS[k] += tau;
            a1[k] += tau * xv;
            a3[k] += tau * tau * tau * x2;
        }
    }

    const int pb = b * TC_ + tc;
#pragma unroll
    for (int k = 0; k < K_; ++k) {
        Spart[((size_t)pb * K_ + k) * D_ + d]                 = aS[k];
        A1p[(((size_t)k * B_ + b) * TC_ + tc) * D_ + d]       = a1[k];
        A3p[(((size_t)k * B_ + b) * TC_ + tc) * D_ + d]       = a3[k];
    }
}

// ---- kernel 3: fixed-order reduction of S partials -> invS -----------------
__global__ void k_reduceS(const float* __restrict__ Spart,
                          float* __restrict__ invS) {
    int i = blockIdx.x * blockDim.x + threadIdx.x;   // < K*D
    float s = 0.0f;
    for (int p = 0; p < B_ * TC_; ++p)
        s += Spart[(size_t)p * (K_ * D_) + i];
    invS[i] = __builtin_amdgcn_rcpf(s + EPS_);
}

// ---- kernel 4: WMMA reduction of A1/A3 t-chunk partials --------------------
// One wave handles one (k, b, 16-wide d-tile). For fixed (k,b) the partials
// form P[16 tc][16 d]; C(16x16) = sum_g ones(16x4) x P[4g..4g+3][.] gives
// column sums (all rows of C identical). f32 in / f32 accumulate == exact.
// EXEC must be all 1s: no divergence before the WMMAs.
__global__ void k_reduceA(const float* __restrict__ A1p,
                          const float* __restrict__ A3p,
                          const float* __restrict__ invS,
                          float* __restrict__ Mco,    // [K][B][D]
                          float* __restrict__ Rco) {  // [K][B][D]
    const int wave = blockIdx.x * (blockDim.x >> 5) + (threadIdx.x >> 5);
    const int lane = threadIdx.x & 31;
    // wave in [0, K*B*(D/16)) = 2048
    const int dt = wave & 15;
    const int b  = (wave >> 4) & 15;
    const int k  = wave >> 8;
    const int col = lane & 15;
    const int hi  = lane >> 4;   // 0: rows {0,1}, 1: rows {2,3} per VGPR pair

    const float* P1 = A1p + ((size_t)k * B_ + b) * TC_ * D_ + dt * 16 + col;
    const float* P3 = A3p + ((size_t)k * B_ + b) * TC_ * D_ + dt * 16 + col;

    v2f aones = {1.0f, 1.0f};
    v8f c1v = {};
    v8f c3v = {};
#pragma unroll
    for (int g = 0; g < 4; ++g) {
        // B-matrix f32 layout: VGPR0 = rows {4g+0 (lanes 0-15), 4g+2 (16-31)},
        //                      VGPR1 = rows {4g+1, 4g+3}; row striped over lanes.
        v2f b1 = { P1[(size_t)(4 * g + 2 * hi + 0) * D_],
                   P1[(size_t)(4 * g + 2 * hi + 1) * D_] };
        v2f b3 = { P3[(size_t)(4 * g + 2 * hi + 0) * D_],
                   P3[(size_t)(4 * g + 2 * hi + 1) * D_] };
        c1v = __builtin_amdgcn_wmma_f32_16x16x4_f32(
                  false, aones, false, b1, (short)0, c1v, false, false);
        c3v = __builtin_amdgcn_wmma_f32_16x16x4_f32(
                  false, aones, false, b3, (short)0, c3v, false, false);
    }

    if (lane < 16) {                      // C VGPR0, M=0 row: lanes 0-15 = cols
        const int d  = dt * 16 + col;
        const float is   = invS[k * D_ + d];
        const float invT = 1.0f / (float)T_;
        const float E1   = c1v[0] * is * invT;                 // mean_t hat_tau*x
        const float E2   = c3v[0] * is * is * is * invT;       // mean_t ht^3*x^2
        const float rstd = rsqrtf(E2 + EPS_);
        const float sc   = rsqrtf(1.0f + EPS_);                // 1/sqrt(pri+eps), pri==1
        const float Rv   = rstd * sc;
        Mco[((size_t)k * B_ + b) * D_ + d] = E1 * Rv;
        Rco[((size_t)k * B_ + b) * D_ + d] = Rv;
    }
}

// ---- kernel 5: streaming pass C -------------------------------------------
// out[b,t,d] = sum_k tau_k * (x * R_k - M_k)
__global__ void k_passC(const float* __restrict__ x,
                        const float* __restrict__ mean,
                        const float* __restrict__ c1,
                        const float* __restrict__ Mco,
                        const float* __restrict__ Rco,
                        float* __restrict__ out) {
    const int d  = threadIdx.x;
    const int tc = blockIdx.x;
    const int b  = blockIdx.y;

    float mk[K_], ck[K_], Mk[K_], Rk[K_];
#pragma unroll
    for (int k = 0; k < K_; ++k) {
        mk[k] = mean[k * D_ + d];
        ck[k] = c1[k * D_ + d];
        Mk[k] = Mco[((size_t)k * B_ + b) * D_ + d];
        Rk[k] = Rco[((size_t)k * B_ + b) * D_ + d];
    }

    const size_t total = (size_t)B_ * T_ * D_;
    const size_t base  = ((size_t)b * T_ + (size_t)tc * CT_) * D_ + d;

    for (int t = 0; t < CT_; ++t) {
        const size_t idx = base + (size_t)t * D_;
        const size_t pf  = idx + 16 * (size_t)D_;
        if (pf < total) __builtin_prefetch(x + pf, 0, 0);

        const float xv = x[idx];
        float p[K_];
        float s = 0.0f;
#pragma unroll
        for (int k = 0; k < K_; ++k) {
            const float dd = xv - mk[k];
            p[k] = __builtin_amdgcn_exp2f(ck[k] * dd * dd);
            s += p[k];
        }
        const float inv = __builtin_amdgcn_rcpf(s + EPS_);
        float acc = 0.0f;
#pragma unroll
        for (int k = 0; k < K_; ++k)
            acc += (p[k] * inv) * (xv * Rk[k] - Mk[k]);
        out[idx] = acc;
    }
}

// ---------------------------------------------------------------------------
extern "C" void kernel_launch(void* const* d_in, const int* in_sizes, int n_in,
                              void* d_out, int out_size, void* d_ws, size_t ws_size,
                              hipStream_t stream) {
    (void)in_sizes; (void)n_in; (void)out_size; (void)ws_size;
    const float* x        = (const float*)d_in[0];   // (B,T,D)
    const float* mean     = (const float*)d_in[1];   // (K,D)
    const float* variance = (const float*)d_in[2];   // (K,D)
    // d_in[3] = prior (K,1): softmax over its singleton axis == 1.0 -> unused.
    float* out = (float*)d_out;

    // Workspace carve-up (floats); total ~6.6 MB.
    float* ws    = (float*)d_ws;
    float* c1    = ws;                                   // K*D
    float* invS  = c1   + (size_t)K_ * D_;               // K*D
    float* Spart = invS + (size_t)K_ * D_;               // (B*TC)*K*D
    float* A1p   = Spart + (size_t)B_ * TC_ * K_ * D_;   // K*B*TC*D
    float* A3p   = A1p   + (size_t)K_ * B_ * TC_ * D_;   // K*B*TC*D
    float* Mco   = A3p   + (size_t)K_ * B_ * TC_ * D_;   // K*B*D
    float* Rco   = Mco   + (size_t)K_ * B_ * D_;         // K*B*D

    k_precompute<<<(K_ * D_) / 256, 256, 0, stream>>>(variance, c1);
    k_passA<<<dim3(TC_, B_), 256, 0, stream>>>(x, mean, c1, Spart, A1p, A3p);
    k_reduceS<<<(K_ * D_) / 256, 256, 0, stream>>>(Spart, invS);
    // 2048 waves, 8 waves (256 threads) per block -> 256 blocks
    k_reduceA<<<(K_ * B_ * (D_ / 16)) / 8, 256, 0, stream>>>(A1p, A3p, invS, Mco, Rco);
    k_passC<<<dim3(TC_, B_), 256, 0, stream>>>(x, mean, c1, Mco, Rco, out);
}